// ElevationAwareGAT_90726889161442
// MI455X (gfx1250) — compile-verified
//
#include <hip/hip_runtime.h>
#include <math.h>

// ---------------- problem constants ----------------
constexpr int NN   = 50000;    // nodes
constexpr int EE   = 400000;   // edges
constexpr int ETOT = EE + NN;  // edges + self loops = 450000
constexpr int HC   = 256;      // heads*channels layer1
constexpr int CC   = 64;       // channels

// ---------------- WMMA helpers ----------------
typedef __attribute__((ext_vector_type(2))) float v2f;
typedef __attribute__((ext_vector_type(8))) float v8f;

__device__ __forceinline__ v8f wmma4(v2f a, v2f b, v8f c) {
  // D = A(16x4,f32) * B(4x16,f32) + C(16x16,f32)
  return __builtin_amdgcn_wmma_f32_16x16x4_f32(false, a, false, b, (short)0, c,
                                               false, false);
}

__device__ __forceinline__ void atomicMaxFloat(float* addr, float val) {
  val += 0.0f;  // canonicalize -0 -> +0 so the sign-split trick is safe
  if (val >= 0.0f)
    atomicMax((int*)addr, __float_as_int(val));
  else
    atomicMin((unsigned int*)addr, __float_as_uint(val));
}

// ---------------- utility kernels ----------------
__global__ void k_fill(float* __restrict__ p, size_t n, float val) {
  size_t i = (size_t)blockIdx.x * 256 + threadIdx.x;
  if (i < n) p[i] = val;
}

__global__ void k_ea_sum(const float* __restrict__ edge_attr, float* __restrict__ acc) {
  float loc[8] = {0, 0, 0, 0, 0, 0, 0, 0};
  int gid = blockIdx.x * blockDim.x + threadIdx.x;
  int stride = gridDim.x * blockDim.x;
  for (int e = gid; e < EE; e += stride) {
#pragma unroll
    for (int k = 0; k < 8; ++k) loc[k] += edge_attr[(size_t)e * 8 + k];
  }
#pragma unroll
  for (int k = 0; k < 8; ++k) {
    float v = loc[k];
    v += __shfl_xor(v, 1); v += __shfl_xor(v, 2); v += __shfl_xor(v, 4);
    v += __shfl_xor(v, 8); v += __shfl_xor(v, 16);
    if ((threadIdx.x & 31) == 0) atomicAdd(&acc[k], v);
  }
}

__global__ void k_ea_div(const float* __restrict__ acc, float* __restrict__ mean) {
  int k = threadIdx.x;
  if (k < 8) mean[k] = acc[k] * (1.0f / (float)EE);
}

// ---------------- layer-1 node transforms (WMMA) ----------------
// xl1 = x@Wl1+bl1 ; xr1 = x@Wr1+br1 ; res = x@Wres   ([N,16]x[16,256] each)
__global__ __launch_bounds__(256) void k_gemm_node1(
    const float* __restrict__ x, const float* __restrict__ Wl,
    const float* __restrict__ bl, const float* __restrict__ Wr,
    const float* __restrict__ br, const float* __restrict__ Wres,
    float* __restrict__ xl1, float* __restrict__ xr1, float* __restrict__ res) {
  __shared__ float A[16 * 17];  // stride 17: conflict-free column reads
  int tid = threadIdx.x;
  int m0 = blockIdx.x * 16;
  {
    int r = tid >> 4, c = tid & 15;
    A[r * 17 + c] = x[(size_t)(m0 + r) * 16 + c];
  }
  __syncthreads();
  int wave = tid >> 5, lane = tid & 31;
  int n = lane & 15;
  int khi = (lane >= 16) ? 2 : 0;
  int rbase = (lane >= 16) ? 8 : 0;
  for (int ct = wave; ct < 48; ct += 8) {
    int gid = ct >> 4;              // 0:Wl1  1:Wr1  2:Wres
    int c0 = (ct & 15) << 4;
    const float* W = (gid == 0) ? Wl : (gid == 1) ? Wr : Wres;
    const float* B = (gid == 0) ? bl : (gid == 1) ? br : nullptr;
    float* O = (gid == 0) ? xl1 : (gid == 1) ? xr1 : res;
    v8f acc = {0, 0, 0, 0, 0, 0, 0, 0};
#pragma unroll
    for (int kk = 0; kk < 4; ++kk) {
      int kb = kk * 4 + khi;
      v2f a, b;
      a.x = A[n * 17 + kb];     a.y = A[n * 17 + kb + 1];
      b.x = W[kb * HC + c0 + n]; b.y = W[(kb + 1) * HC + c0 + n];
      acc = wmma4(a, b, acc);
    }
    float bias = B ? B[c0 + n] : 0.0f;
#pragma unroll
    for (int r = 0; r < 8; ++r)
      O[(size_t)(m0 + rbase + r) * HC + c0 + n] = acc[r] + bias;
  }
}

// ---------------- layer-1 edge attention logits ----------------
__global__ __launch_bounds__(256) void k_edge_alpha1(
    const int* __restrict__ row, const int* __restrict__ col,
    const float* __restrict__ edge_attr, const float* __restrict__ ea_mean,
    const float* __restrict__ We, const float* __restrict__ att,
    const float* __restrict__ xl1, const float* __restrict__ xr1,
    float* __restrict__ alpha, float* __restrict__ nmax) {
  __shared__ float sWe[8 * HC];
  __shared__ float sAtt[HC];
  int tid = threadIdx.x;
  for (int i = tid; i < 8 * HC; i += 256) sWe[i] = We[i];
  for (int i = tid; i < HC; i += 256) sAtt[i] = att[i];
  __syncthreads();
  int wave = tid >> 5, lane = tid & 31;
  int e = blockIdx.x * 8 + wave;  // ETOT divisible by 8
  int s, d;
  const float* ea;
  if (e < EE) { s = row[e]; d = col[e]; ea = edge_attr + (size_t)e * 8; }
  else        { s = e - EE; d = s;      ea = ea_mean; }
  float eav[8];
#pragma unroll
  for (int k = 0; k < 8; ++k) eav[k] = ea[k];
  const float* xls = xl1 + (size_t)s * HC;
  const float* xrd = xr1 + (size_t)d * HC;
  float hs[4] = {0, 0, 0, 0};
#pragma unroll
  for (int t = 0; t < 8; ++t) {
    int j = lane + 32 * t;                 // coalesced gathers
    float v = xls[j] + xrd[j];
    float ee = 0.0f;
#pragma unroll
    for (int k = 0; k < 8; ++k) ee += eav[k] * sWe[k * HC + j];
    v += ee;
    v = v > 0.0f ? v : 0.2f * v;           // leaky_relu 0.2
    hs[t >> 1] += v * sAtt[j];
  }
#pragma unroll
  for (int h = 0; h < 4; ++h) {
    float v = hs[h];
    v += __shfl_xor(v, 1); v += __shfl_xor(v, 2); v += __shfl_xor(v, 4);
    v += __shfl_xor(v, 8); v += __shfl_xor(v, 16);
    hs[h] = v;
  }
  if (lane == 0) {
#pragma unroll
    for (int h = 0; h < 4; ++h) {
      alpha[(size_t)e * 4 + h] = hs[h];
      atomicMaxFloat(&nmax[(size_t)d * 4 + h], hs[h]);
    }
  }
}

__global__ void k_softmax1(const int* __restrict__ col,
                           const float* __restrict__ nmax,
                           float* __restrict__ alpha, float* __restrict__ nsum) {
  int i = blockIdx.x * 256 + threadIdx.x;
  if (i >= ETOT * 4) return;
  int e = i >> 2, h = i & 3;
  int d = (e < EE) ? col[e] : e - EE;
  float ex = expf(alpha[i] - nmax[(size_t)d * 4 + h]);
  alpha[i] = ex;
  atomicAdd(&nsum[(size_t)d * 4 + h], ex);
}

__global__ __launch_bounds__(256) void k_agg1(
    const int* __restrict__ row, const int* __restrict__ col,
    const float* __restrict__ xl1, const float* __restrict__ alpha,
    const float* __restrict__ nsum, float* __restrict__ agg) {
  int tid = threadIdx.x, wave = tid >> 5, lane = tid & 31;
  int e = blockIdx.x * 8 + wave;
  int s, d;
  if (e < EE) { s = row[e]; d = col[e]; } else { s = e - EE; d = s; }
  float a[4];
#pragma unroll
  for (int h = 0; h < 4; ++h)
    a[h] = alpha[(size_t)e * 4 + h] / (nsum[(size_t)d * 4 + h] + 1e-16f);
  const float* xs = xl1 + (size_t)s * HC;
  float* ag = agg + (size_t)d * HC;
#pragma unroll
  for (int t = 0; t < 8; ++t) {
    int j = lane + 32 * t;
    atomicAdd(&ag[j], xs[j] * a[t >> 1]);
  }
}

// bias1 + bn1 + residual + elu  (in place: agg -> h1)
__global__ void k_post1(float* __restrict__ agg, const float* __restrict__ res,
                        const float* __restrict__ bias1,
                        const float* __restrict__ g, const float* __restrict__ b,
                        const float* __restrict__ m, const float* __restrict__ v) {
  size_t i = (size_t)blockIdx.x * 256 + threadIdx.x;  // N*HC exact
  int j = (int)(i & 255);
  float h = agg[i] + bias1[j];
  h = (h - m[j]) * rsqrtf(v[j] + 1e-5f) * g[j] + b[j];
  h += res[i];
  agg[i] = h > 0.0f ? h : expm1f(h);
}

// ---------------- layer-2 node transforms (WMMA) ----------------
// xl2 = h1@Wl2+bl2 ; xr2 = h1@Wr2+br2   ([N,256]x[256,64])
__global__ __launch_bounds__(256) void k_gemm_node2(
    const float* __restrict__ h1, const float* __restrict__ Wl,
    const float* __restrict__ bl, const float* __restrict__ Wr,
    const float* __restrict__ br, float* __restrict__ xl2,
    float* __restrict__ xr2) {
  __shared__ float A[16 * 259];  // stride 259 (==3 mod 64): conflict-free
  int tid = threadIdx.x;
  int m0 = blockIdx.x * 16;
  for (int idx = tid; idx < 16 * HC; idx += 256) {
    int r = idx >> 8, c = idx & 255;
    A[r * 259 + c] = h1[(size_t)(m0 + r) * HC + c];
  }
  __syncthreads();
  int wave = tid >> 5, lane = tid & 31;
  int n = lane & 15, khi = (lane >= 16) ? 2 : 0, rbase = (lane >= 16) ? 8 : 0;
  int gid = wave >> 2;          // 0: Wl2, 1: Wr2
  int c0 = (wave & 3) << 4;
  const float* W = gid ? Wr : Wl;
  const float* B = gid ? br : bl;
  float* O = gid ? xr2 : xl2;
  v8f acc = {0, 0, 0, 0, 0, 0, 0, 0};
  for (int kk = 0; kk < 64; ++kk) {
    int kb = kk * 4 + khi;
    v2f a, b;
    a.x = A[n * 259 + kb];     a.y = A[n * 259 + kb + 1];
    b.x = W[kb * CC + c0 + n]; b.y = W[(kb + 1) * CC + c0 + n];
    acc = wmma4(a, b, acc);
  }
  float bias = B[c0 + n];
#pragma unroll
  for (int r = 0; r < 8; ++r)
    O[(size_t)(m0 + rbase + r) * CC + c0 + n] = acc[r] + bias;
}

// ---------------- layer-2 edge attention ----------------
__global__ __launch_bounds__(256) void k_edge_alpha2(
    const int* __restrict__ row, const int* __restrict__ col,
    const float* __restrict__ edge_attr, const float* __restrict__ ea_mean,
    const float* __restrict__ We, const float* __restrict__ att,
    const float* __restrict__ xl2, const float* __restrict__ xr2,
    float* __restrict__ alpha, float* __restrict__ nmax) {
  __shared__ float sWe[8 * CC];
  __shared__ float sAtt[CC];
  int tid = threadIdx.x;
  for (int i = tid; i < 8 * CC; i += 256) sWe[i] = We[i];
  for (int i = tid; i < CC; i += 256) sAtt[i] = att[i];
  __syncthreads();
  int wave = tid >> 5, lane = tid & 31;
  int e = blockIdx.x * 8 + wave;
  int s, d;
  const float* ea;
  if (e < EE) { s = row[e]; d = col[e]; ea = edge_attr + (size_t)e * 8; }
  else        { s = e - EE; d = s;      ea = ea_mean; }
  float eav[8];
#pragma unroll
  for (int k = 0; k < 8; ++k) eav[k] = ea[k];
  const float* xls = xl2 + (size_t)s * CC;
  const float* xrd = xr2 + (size_t)d * CC;
  float sum = 0.0f;
#pragma unroll
  for (int t = 0; t < 2; ++t) {
    int j = lane + 32 * t;
    float v = xls[j] + xrd[j];
    float ee = 0.0f;
#pragma unroll
    for (int k = 0; k < 8; ++k) ee += eav[k] * sWe[k * CC + j];
    v += ee;
    v = v > 0.0f ? v : 0.2f * v;
    sum += v * sAtt[j];
  }
  sum += __shfl_xor(sum, 1); sum += __shfl_xor(sum, 2); sum += __shfl_xor(sum, 4);
  sum += __shfl_xor(sum, 8); sum += __shfl_xor(sum, 16);
  if (lane == 0) {
    alpha[e] = sum;
    atomicMaxFloat(&nmax[d], sum);
  }
}

__global__ void k_softmax2(const int* __restrict__ col,
                           const float* __restrict__ nmax,
                           float* __restrict__ alpha, float* __restrict__ nsum) {
  int e = blockIdx.x * 256 + threadIdx.x;
  if (e >= ETOT) return;
  int d = (e < EE) ? col[e] : e - EE;
  float ex = expf(alpha[e] - nmax[d]);
  alpha[e] = ex;
  atomicAdd(&nsum[d], ex);
}

__global__ __launch_bounds__(256) void k_agg2(
    const int* __restrict__ row, const int* __restrict__ col,
    const float* __restrict__ xl2, const float* __restrict__ alpha,
    const float* __restrict__ nsum, float* __restrict__ agg) {
  int tid = threadIdx.x, wave = tid >> 5, lane = tid & 31;
  int e = blockIdx.x * 8 + wave;
  int s, d;
  if (e < EE) { s = row[e]; d = col[e]; } else { s = e - EE; d = s; }
  float a = alpha[e] / (nsum[d] + 1e-16f);
  const float* xs = xl2 + (size_t)s * CC;
  float* ag = agg + (size_t)d * CC;
#pragma unroll
  for (int t = 0; t < 2; ++t) {
    int j = lane + 32 * t;
    atomicAdd(&ag[j], xs[j] * a);
  }
}

// bias2 + bn2 + elu (in place: agg -> h2)
__global__ void k_post2(float* __restrict__ agg, const float* __restrict__ bias2,
                        const float* __restrict__ g, const float* __restrict__ b,
                        const float* __restrict__ m, const float* __restrict__ v) {
  size_t i = (size_t)blockIdx.x * 256 + threadIdx.x;  // N*CC exact
  int j = (int)(i & 63);
  float h = agg[i] + bias2[j];
  h = (h - m[j]) * rsqrtf(v[j] + 1e-5f) * g[j] + b[j];
  agg[i] = h > 0.0f ? h : expm1f(h);
}

// ---------------- fused edge MLP head (2 WMMA stages + dot product + sigmoid) ----------------
__device__ __forceinline__ float gelu_exact(float x) {
  return 0.5f * x * (1.0f + erff(x * 0.70710678118654752f));
}

__global__ __launch_bounds__(128) void k_edge_mlp(
    const int* __restrict__ row, const int* __restrict__ col,
    const float* __restrict__ edge_attr, const float* __restrict__ h2,
    const float* __restrict__ Wm1, const float* __restrict__ bm1,
    const float* __restrict__ Wm2, const float* __restrict__ bm2,
    const float* __restrict__ Wm3, const float* __restrict__ bm3,
    float* __restrict__ out) {
  __shared__ float ER[16 * 137];  // [16,136] gathered features, stride 137
  __shared__ float Z1[16 * 67];   // [16,64] hidden1, stride 67
  __shared__ float Z2[16 * 35];   // [16,32] hidden2, stride 35
  int tid = threadIdx.x;
  int e0 = blockIdx.x * 16;       // EE divisible by 16
  for (int idx = tid; idx < 16 * 136; idx += 128) {
    int el = idx / 136, j = idx - el * 136;
    int e = e0 + el;
    float v;
    if (j < 64)       v = h2[(size_t)row[e] * CC + j];
    else if (j < 128) v = h2[(size_t)col[e] * CC + (j - 64)];
    else              v = edge_attr[(size_t)e * 8 + (j - 128)];
    ER[el * 137 + j] = v;
  }
  __syncthreads();
  int wave = tid >> 5, lane = tid & 31;
  int n = lane & 15, khi = (lane >= 16) ? 2 : 0, rbase = (lane >= 16) ? 8 : 0;
  // stage 1: [16,136] x Wm1[136,64] -> gelu -> Z1
  {
    int c0 = wave * 16;
    v8f acc = {0, 0, 0, 0, 0, 0, 0, 0};
    for (int kk = 0; kk < 34; ++kk) {
      int kb = kk * 4 + khi;
      v2f a, b;
      a.x = ER[n * 137 + kb];      a.y = ER[n * 137 + kb + 1];
      b.x = Wm1[kb * 64 + c0 + n]; b.y = Wm1[(kb + 1) * 64 + c0 + n];
      acc = wmma4(a, b, acc);
    }
    float bias = bm1[c0 + n];
#pragma unroll
    for (int r = 0; r < 8; ++r)
      Z1[(rbase + r) * 67 + c0 + n] = gelu_exact(acc[r] + bias);
  }
  __syncthreads();
  // stage 2: [16,64] x Wm2[64,32] -> gelu -> Z2   (waves 0,1)
  if (wave < 2) {
    int c0 = wave * 16;
    v8f acc = {0, 0, 0, 0, 0, 0, 0, 0};
    for (int kk = 0; kk < 16; ++kk) {
      int kb = kk * 4 + khi;
      v2f a, b;
      a.x = Z1[n * 67 + kb];       a.y = Z1[n * 67 + kb + 1];
      b.x = Wm2[kb * 32 + c0 + n]; b.y = Wm2[(kb + 1) * 32 + c0 + n];
      acc = wmma4(a, b, acc);
    }
    float bias = bm2[c0 + n];
#pragma unroll
    for (int r = 0; r < 8; ++r)
      Z2[(rbase + r) * 35 + c0 + n] = gelu_exact(acc[r] + bias);
  }
  __syncthreads();
  // stage 3: [16,32] x Wm3[32,1] + sigmoid
  if (tid < 16) {
    float s = bm3[0];
#pragma unroll
    for (int k = 0; k < 32; ++k) s += Z2[tid * 35 + k] * Wm3[k];
    out[e0 + tid] = 1.0f / (1.0f + expf(-s));
  }
}

// ---------------- host launch ----------------
extern "C" void kernel_launch(void* const* d_in, const int* in_sizes, int n_in,
                              void* d_out, int out_size, void* d_ws, size_t ws_size,
                              hipStream_t stream) {
  const float* x     = (const float*)d_in[0];
  const int* eidx    = (const int*)d_in[1];
  const float* eattr = (const float*)d_in[2];
  const float* Wl1 = (const float*)d_in[3],  *bl1 = (const float*)d_in[4];
  const float* Wr1 = (const float*)d_in[5],  *br1 = (const float*)d_in[6];
  const float* We1 = (const float*)d_in[7],  *att1 = (const float*)d_in[8];
  const float* bias1 = (const float*)d_in[9];
  const float* Wl2 = (const float*)d_in[10], *bl2 = (const float*)d_in[11];
  const float* Wr2 = (const float*)d_in[12], *br2 = (const float*)d_in[13];
  const float* We2 = (const float*)d_in[14], *att2 = (const float*)d_in[15];
  const float* bias2 = (const float*)d_in[16];
  const float* bn1_g = (const float*)d_in[17], *bn1_b = (const float*)d_in[18];
  const float* bn1_m = (const float*)d_in[19], *bn1_v = (const float*)d_in[20];
  const float* bn2_g = (const float*)d_in[21], *bn2_b = (const float*)d_in[22];
  const float* bn2_m = (const float*)d_in[23], *bn2_v = (const float*)d_in[24];
  const float* Wres = (const float*)d_in[25];
  const float* Wm1 = (const float*)d_in[26], *bm1 = (const float*)d_in[27];
  const float* Wm2 = (const float*)d_in[28], *bm2 = (const float*)d_in[29];
  const float* Wm3 = (const float*)d_in[30], *bm3 = (const float*)d_in[31];
  const int* row = eidx;
  const int* col = eidx + EE;

  float* ws = (float*)d_ws;
  const size_t oXL1  = 0;
  const size_t oXR1  = oXL1 + (size_t)NN * HC;
  const size_t oRES  = oXR1 + (size_t)NN * HC;
  const size_t oAGG1 = oRES + (size_t)NN * HC;   // becomes h1 in place
  const size_t oAL1  = oAGG1 + (size_t)NN * HC;
  const size_t oNM1  = oAL1 + (size_t)ETOT * 4;
  const size_t oNS1  = oNM1 + (size_t)NN * 4;
  const size_t oXL2  = oNS1 + (size_t)NN * 4;
  const size_t oXR2  = oXL2 + (size_t)NN * CC;
  const size_t oAGG2 = oXR2 + (size_t)NN * CC;   // becomes h2 in place
  const size_t oAL2  = oAGG2 + (size_t)NN * CC;
  const size_t oNM2  = oAL2 + (size_t)ETOT;
  const size_t oNS2  = oNM2 + (size_t)NN;
  const size_t oEAC  = oNS2 + (size_t)NN;
  const size_t oEAM  = oEAC + 8;

  const float NEGINF = -INFINITY;
  auto fill = [&](size_t off, size_t n, float v) {
    k_fill<<<(unsigned)((n + 255) / 256), 256, 0, stream>>>(ws + off, n, v);
  };
  fill(oAGG1, (size_t)NN * HC, 0.0f);
  fill(oNM1, (size_t)NN * 4, NEGINF);
  fill(oNS1, (size_t)NN * 4, 0.0f);
  fill(oAGG2, (size_t)NN * CC, 0.0f);
  fill(oNM2, (size_t)NN, NEGINF);
  fill(oNS2, (size_t)NN, 0.0f);
  fill(oEAC, 8, 0.0f);

  k_ea_sum<<<120, 256, 0, stream>>>(eattr, ws + oEAC);
  k_ea_div<<<1, 32, 0, stream>>>(ws + oEAC, ws + oEAM);

  // ---- layer 1 ----
  k_gemm_node1<<<NN / 16, 256, 0, stream>>>(x, Wl1, bl1, Wr1, br1, Wres,
                                            ws + oXL1, ws + oXR1, ws + oRES);
  k_edge_alpha1<<<ETOT / 8, 256, 0, stream>>>(row, col, eattr, ws + oEAM, We1,
                                              att1, ws + oXL1, ws + oXR1,
                                              ws + oAL1, ws + oNM1);
  k_softmax1<<<(ETOT * 4 + 255) / 256, 256, 0, stream>>>(col, ws + oNM1,
                                                         ws + oAL1, ws + oNS1);
  k_agg1<<<ETOT / 8, 256, 0, stream>>>(row, col, ws + oXL1, ws + oAL1,
                                       ws + oNS1, ws + oAGG1);
  k_post1<<<(unsigned)((size_t)NN * HC / 256), 256, 0, stream>>>(
      ws + oAGG1, ws + oRES, bias1, bn1_g, bn1_b, bn1_m, bn1_v);

  // ---- layer 2 ----
  k_gemm_node2<<<NN / 16, 256, 0, stream>>>(ws + oAGG1, Wl2, bl2, Wr2, br2,
                                            ws + oXL2, ws + oXR2);
  k_edge_alpha2<<<ETOT / 8, 256, 0, stream>>>(row, col, eattr, ws + oEAM, We2,
                                              att2, ws + oXL2, ws + oXR2,
                                              ws + oAL2, ws + oNM2);
  k_softmax2<<<(ETOT + 255) / 256, 256, 0, stream>>>(col, ws + oNM2, ws + oAL2,
                                                     ws + oNS2);
  k_agg2<<<ETOT / 8, 256, 0, stream>>>(row, col, ws + oXL2, ws + oAL2,
                                       ws + oNS2, ws + oAGG2);
  k_post2<<<(unsigned)((size_t)NN * CC / 256), 256, 0, stream>>>(
      ws + oAGG2, bias2, bn2_g, bn2_b, bn2_m, bn2_v);

  // ---- edge regressor head ----
  k_edge_mlp<<<EE / 16, 128, 0, stream>>>(row, col, eattr, ws + oAGG2, Wm1, bm1,
                                          Wm2, bm2, Wm3, bm3, (float*)d_out);
}